// NMPEncoder_67061619359862
// MI455X (gfx1250) — compile-verified
//
#include <hip/hip_runtime.h>
#include <hip/hip_bf16.h>

// ---------------- problem constants ----------------
#define Bq   16
#define Nq   80
#define Eq   6320          // N*(N-1)
#define NM1  79            // N-1
#define NHq  256           // NHID
#define VISq 4096
#define SEMq 300
#define INW  4396          // VIS+SEM row width of `inputs`
#define RN   1280          // B*N node rows
#define RE   101120        // B*E edge rows
#define EDGE_T 71
#define NODE_T 101

typedef __attribute__((ext_vector_type(2))) float v2f;
typedef __attribute__((ext_vector_type(8))) float v8f;

enum { ACT_NONE = 0, ACT_ELU = 1, ACT_RELU = 2 };
enum { AM_PLAIN = 0, AM_N2E = 1, AM_M3 = 2 };

__device__ __forceinline__ v8f wmma_f32(v2f a, v2f b, v8f c) {
  return __builtin_amdgcn_wmma_f32_16x16x4_f32(false, a, false, b, (short)0, c,
                                               false, false);
}

// One K-segment of STEPS k-steps (4 K-values each) accumulating a 16x64 tile.
// FULL: all 4 column-tiles in range -> single B pointer, immediate offsets.
template <int NOUT, int STEPS, bool FULL>
__device__ __forceinline__ void gemm_seg(const float* __restrict__ ap,
                                         const float* __restrict__ bp0,
                                         const float* __restrict__ bp1,
                                         const float* __restrict__ bp2,
                                         const float* __restrict__ bp3,
                                         v8f* acc)
{
  const size_t bstep = 4 * (size_t)NOUT;
#pragma unroll 4
  for (int i = 0; i < STEPS; ++i) {
    const v2f a = *(const v2f*)ap;                 // 8B-aligned b64 load
    v2f b0, b1, b2, b3;
    if (FULL) {
      b0.x = bp0[0];       b0.y = bp0[NOUT];
      b1.x = bp0[16];      b1.y = bp0[16 + NOUT];
      b2.x = bp0[32];      b2.y = bp0[32 + NOUT];
      b3.x = bp0[48];      b3.y = bp0[48 + NOUT];
      bp0 += bstep;
    } else {
      b0.x = bp0[0];       b0.y = bp0[NOUT];
      b1.x = bp1[0];       b1.y = bp1[NOUT];
      b2.x = bp2[0];       b2.y = bp2[NOUT];
      b3.x = bp3[0];       b3.y = bp3[NOUT];
      bp0 += bstep; bp1 += bstep; bp2 += bstep; bp3 += bstep;
    }
    acc[0] = wmma_f32(a, b0, acc[0]);
    acc[1] = wmma_f32(a, b1, acc[1]);
    acc[2] = wmma_f32(a, b2, acc[2]);
    acc[3] = wmma_f32(a, b3, acc[3]);
    ap += 4;
  }
}

// ---------------------------------------------------------------------------
// fp32 WMMA GEMM: Y[R,NOUT] = act(A[R,K] @ W[K,NOUT] + bias)
// One wave computes a 16x64 output tile (4 accumulators) with
// V_WMMA_F32_16X16X4_F32. AMODE selects A-row sourcing:
//   AM_PLAIN : A[row,k] = X[row*ldx + k]          (K = KC, compile-time)
//   AM_N2E   : K=512 : [node(recv) | node(send)]
//   AM_M3    : K=768 : [h(recv) | h(send) | edge]
// ---------------------------------------------------------------------------
template <int AMODE, int ACT, int NOUT, int KC>
__global__ __launch_bounds__(32) void wmma_gemm_kernel(
    const float* __restrict__ X, int ldx,
    const float* __restrict__ W, const float* __restrict__ bias,
    float* __restrict__ Y, int ldy,
    const float* __restrict__ nodeSrc, const float* __restrict__ edgeSrc)
{
  const int lane  = threadIdx.x;
  const int laneM = lane & 15;
  const int hi    = lane >> 4;          // 0 -> K offsets {0,1}; 1 -> {2,3}
  const int k0    = hi * 2;
  const int n0    = blockIdx.x * 64;
  const int m0    = blockIdx.y * 16;
  const int col   = n0 + laneM;
  const int row   = m0 + laneM;         // this lane's A row (fixed for the tile)

  const float* aR;
  const float* aS = nullptr;
  const float* aE = nullptr;
  if (AMODE == AM_PLAIN) {
    aR = X + (size_t)row * (size_t)ldx;
  } else {
    const int b  = row / Eq;
    const int e  = row - b * Eq;
    const int r  = e / NM1;
    const int si = e - r * NM1;
    const int s  = si + (si >= r ? 1 : 0);
    aR = nodeSrc + (size_t)(b * Nq + r) * NHq;
    aS = nodeSrc + (size_t)(b * Nq + s) * NHq;
    if (AMODE == AM_M3) aE = edgeSrc + (size_t)row * NHq;
  }

  v8f acc[4] = {};
  const bool full = (n0 + 64 <= NOUT);  // uniform

  if (full) {
    const float* wb = W + (size_t)k0 * NOUT + col;
    if (AMODE == AM_PLAIN) {
      gemm_seg<NOUT, KC / 4, true>(aR + k0, wb, nullptr, nullptr, nullptr, acc);
    } else {
      gemm_seg<NOUT, 64, true>(aR + k0, wb, nullptr, nullptr, nullptr, acc);
      gemm_seg<NOUT, 64, true>(aS + k0, wb + (size_t)256 * NOUT, nullptr, nullptr,
                               nullptr, acc);
      if (AMODE == AM_M3)
        gemm_seg<NOUT, 64, true>(aE + k0, wb + (size_t)512 * NOUT, nullptr,
                                 nullptr, nullptr, acc);
    }
  } else {
    // clamp each column tile (junk columns never stored)
    const int c0 = (col      < NOUT) ? col      : NOUT - 1;
    const int c1 = (col + 16 < NOUT) ? col + 16 : NOUT - 1;
    const int c2 = (col + 32 < NOUT) ? col + 32 : NOUT - 1;
    const int c3 = (col + 48 < NOUT) ? col + 48 : NOUT - 1;
    const float* w0 = W + (size_t)k0 * NOUT + c0;
    const float* w1 = W + (size_t)k0 * NOUT + c1;
    const float* w2 = W + (size_t)k0 * NOUT + c2;
    const float* w3 = W + (size_t)k0 * NOUT + c3;
    if (AMODE == AM_PLAIN) {
      gemm_seg<NOUT, KC / 4, false>(aR + k0, w0, w1, w2, w3, acc);
    } else {
      const size_t so = (size_t)256 * NOUT;
      gemm_seg<NOUT, 64, false>(aR + k0, w0, w1, w2, w3, acc);
      gemm_seg<NOUT, 64, false>(aS + k0, w0 + so, w1 + so, w2 + so, w3 + so, acc);
      if (AMODE == AM_M3)
        gemm_seg<NOUT, 64, false>(aE + k0, w0 + 2 * so, w1 + 2 * so, w2 + 2 * so,
                                  w3 + 2 * so, acc);
    }
  }

  // epilogue: C/D layout -> lanes 0-15: M=v ; lanes 16-31: M=8+v
  const int rbase = m0 + hi * 8;
  float* ybase = Y + (size_t)rbase * (size_t)ldy;
#pragma unroll
  for (int t = 0; t < 4; ++t) {
    const int c = col + 16 * t;
    if (c < NOUT) {
      const float bb = bias[c];
      float* yp = ybase + c;
#pragma unroll
      for (int v = 0; v < 8; ++v) {
        float val = acc[t][v] + bb;
        if (ACT == ACT_ELU)       val = (val > 0.0f) ? val : (__expf(val) - 1.0f);
        else if (ACT == ACT_RELU) val = fmaxf(val, 0.0f);
        yp[(size_t)v * (size_t)ldy] = val;
      }
    }
  }
}

// ---------------------------------------------------------------------------
// BatchNorm (training-mode stats over all R rows)
// ---------------------------------------------------------------------------
__global__ void zero_stats_kernel(float* __restrict__ p) {
  p[threadIdx.x] = 0.0f;
}

__global__ void bn_stats_kernel(const float* __restrict__ X, int ld, int R, int D,
                                float* __restrict__ sum, float* __restrict__ sumsq,
                                int rowsPerBlock)
{
  const int c = threadIdx.x;
  if (c >= D) return;
  int r0 = blockIdx.x * rowsPerBlock;
  int r1 = r0 + rowsPerBlock; if (r1 > R) r1 = R;
  float s = 0.0f, s2 = 0.0f;
  const float* p = X + (size_t)r0 * (size_t)ld + c;
  for (int r = r0; r < r1; ++r) {
    float v = *p;
    s += v; s2 += v * v;
    p += ld;
  }
  atomicAdd(&sum[c], s);
  atomicAdd(&sumsq[c], s2);
}

__global__ void bn_apply_kernel(float* __restrict__ X, int ld, int R, int D,
                                const float* __restrict__ sum,
                                const float* __restrict__ sumsq,
                                const float* __restrict__ g,
                                const float* __restrict__ bt)
{
  const size_t total = (size_t)R * (size_t)D;
  for (size_t idx = (size_t)blockIdx.x * blockDim.x + threadIdx.x; idx < total;
       idx += (size_t)gridDim.x * blockDim.x) {
    const int r = (int)(idx / (size_t)D);
    const int c = (int)(idx - (size_t)r * D);
    const float invR = 1.0f / (float)R;
    const float mu   = sum[c] * invR;
    const float var  = sumsq[c] * invR - mu * mu;
    float* p = X + (size_t)r * (size_t)ld + c;
    *p = (*p - mu) * rsqrtf(var + 1e-5f) * g[c] + bt[c];
  }
}

// ---------------------------------------------------------------------------
// edge2node: Out[B*N, 512] = [mean incoming (c<256) | mean outgoing] of Ein[B*E,256]
// ---------------------------------------------------------------------------
__global__ void edge2node_kernel(const float* __restrict__ Ein, float* __restrict__ Out)
{
  const size_t total = (size_t)RN * 512;
  for (size_t idx = (size_t)blockIdx.x * blockDim.x + threadIdx.x; idx < total;
       idx += (size_t)gridDim.x * blockDim.x) {
    const int row = (int)(idx >> 9);
    const int c   = (int)(idx & 511);
    const int b   = row / Nq;
    const int n   = row - b * Nq;
    float s = 0.0f;
    if (c < 256) {
      // incoming: receiver == n  ->  e in [n*79, n*79+79)
      const float* p = Ein + ((size_t)(b * Eq + n * NM1)) * NHq + c;
      for (int j = 0; j < NM1; ++j) s += p[(size_t)j * NHq];
    } else {
      // outgoing: sender == n  ->  e = r*79 + (n - (n>r)) for all r != n
      const int c2 = c - 256;
      for (int r = 0; r < Nq; ++r) {
        if (r == n) continue;
        const int si = n - (n > r ? 1 : 0);
        const int e  = r * NM1 + si;
        s += Ein[((size_t)(b * Eq + e)) * NHq + c2];
      }
    }
    Out[idx] = s * (1.0f / (float)NM1);
  }
}

// ---------------------------------------------------------------------------
// Host-side helpers
// ---------------------------------------------------------------------------
template <int AMODE, int ACT, int NOUT, int KC>
static inline void launch_gemm(hipStream_t st, const float* X, int ldx,
                               const float* W, const float* b, float* Y, int ldy,
                               int R, const float* nodeSrc, const float* edgeSrc)
{
  dim3 grid((unsigned)((NOUT + 63) / 64), (unsigned)(R / 16), 1);
  wmma_gemm_kernel<AMODE, ACT, NOUT, KC><<<grid, dim3(32), 0, st>>>(
      X, ldx, W, b, Y, ldy, nodeSrc, edgeSrc);
}

static inline void launch_bn(hipStream_t st, float* X, int ld, int R, int D,
                             float* stats, const float* g, const float* bt)
{
  float* sum = stats;
  float* sumsq = stats + 256;
  zero_stats_kernel<<<1, 512, 0, st>>>(stats);
  const int chunks = 128;
  const int rpb = (R + chunks - 1) / chunks;
  bn_stats_kernel<<<chunks, D, 0, st>>>(X, ld, R, D, sum, sumsq, rpb);
  size_t total = (size_t)R * (size_t)D;
  unsigned blocks = (unsigned)((total + 255) / 256);
  if (blocks > 4096u) blocks = 4096u;
  bn_apply_kernel<<<blocks, 256, 0, st>>>(X, ld, R, D, sum, sumsq, g, bt);
}

// ---------------------------------------------------------------------------
extern "C" void kernel_launch(void* const* d_in, const int* in_sizes, int n_in,
                              void* d_out, int out_size, void* d_ws, size_t ws_size,
                              hipStream_t stream)
{
  (void)in_sizes; (void)n_in; (void)out_size; (void)ws_size;

  // setup_inputs() dict order (leaves in insertion order):
  // 0 inputs, 1 spatial_feats(unused), 2 rel_rec(unused), 3 rel_send(unused),
  // 4 bbox_loc(unused), then params: vis{W1,b1,W2,b2,g,bt}, sem{...}, fus{W,b},
  // cls{W,b}, m1{...}, e2n{...}, m2{...}, m3{...}, rel{W,b}
  const float* inputs = (const float*)d_in[0];
  const float* visW1 = (const float*)d_in[5];
  const float* visb1 = (const float*)d_in[6];
  const float* visW2 = (const float*)d_in[7];
  const float* visb2 = (const float*)d_in[8];
  const float* visg  = (const float*)d_in[9];
  const float* visbt = (const float*)d_in[10];
  const float* semW1 = (const float*)d_in[11];
  const float* semb1 = (const float*)d_in[12];
  const float* semW2 = (const float*)d_in[13];
  const float* semb2 = (const float*)d_in[14];
  const float* semg  = (const float*)d_in[15];
  const float* sembt = (const float*)d_in[16];
  const float* fusW  = (const float*)d_in[17];
  const float* fusb  = (const float*)d_in[18];
  const float* clsW  = (const float*)d_in[19];
  const float* clsb  = (const float*)d_in[20];
  const float* m1W1  = (const float*)d_in[21];
  const float* m1b1  = (const float*)d_in[22];
  const float* m1W2  = (const float*)d_in[23];
  const float* m1b2  = (const float*)d_in[24];
  const float* m1g   = (const float*)d_in[25];
  const float* m1bt  = (const float*)d_in[26];
  const float* e2nW1 = (const float*)d_in[27];
  const float* e2nb1 = (const float*)d_in[28];
  const float* e2nW2 = (const float*)d_in[29];
  const float* e2nb2 = (const float*)d_in[30];
  const float* e2ng  = (const float*)d_in[31];
  const float* e2nbt = (const float*)d_in[32];
  const float* m2W1  = (const float*)d_in[33];
  const float* m2b1  = (const float*)d_in[34];
  const float* m2W2  = (const float*)d_in[35];
  const float* m2b2  = (const float*)d_in[36];
  const float* m2g   = (const float*)d_in[37];
  const float* m2bt  = (const float*)d_in[38];
  const float* m3W1  = (const float*)d_in[39];
  const float* m3b1  = (const float*)d_in[40];
  const float* m3W2  = (const float*)d_in[41];
  const float* m3b2  = (const float*)d_in[42];
  const float* m3g   = (const float*)d_in[43];
  const float* m3bt  = (const float*)d_in[44];
  const float* relW  = (const float*)d_in[45];
  const float* relb  = (const float*)d_in[46];

  // workspace layout (floats)
  float* stats  = (float*)d_ws;                       // 512 (sum|sumsq)
  float* nodeA  = stats + 512;                        // [1280,384] vis|sem concat
  float* nodeH  = nodeA  + (size_t)RN * 384;          // [1280,256] hidden tmp
  float* nodeF  = nodeH  + (size_t)RN * 256;          // [1280,256] node_feats / h
  float* nodeG  = nodeF  + (size_t)RN * 256;          // [1280,512] edge2node out
  float* nodeH2 = nodeG  + (size_t)RN * 512;          // [1280,256]
  float* edgeA  = nodeH2 + (size_t)RN * 256;          // [101120,256]
  float* edgeB  = edgeA  + (size_t)RE * 256;          // [101120,256]

  float* out_rel = (float*)d_out;                         // [B,E,71]
  float* out_cls = (float*)d_out + (size_t)RE * EDGE_T;   // [B,N,101]

  // ---- vis MLP: [1280,4096] -> 128 -> 128, BN into nodeA[:, :128] ----
  launch_gemm<AM_PLAIN, ACT_ELU, 128, VISq>(stream, inputs, INW, visW1, visb1,
                                            nodeH, 128, RN, nullptr, nullptr);
  launch_gemm<AM_PLAIN, ACT_ELU, 128, 128>(stream, nodeH, 128, visW2, visb2,
                                           nodeA, 384, RN, nullptr, nullptr);
  launch_bn(stream, nodeA, 384, RN, 128, stats, visg, visbt);

  // ---- sem MLP: [1280,300] -> 256 -> 256, BN into nodeA[:, 128:384] ----
  launch_gemm<AM_PLAIN, ACT_ELU, 256, SEMq>(stream, inputs + VISq, INW, semW1,
                                            semb1, nodeH, 256, RN, nullptr, nullptr);
  launch_gemm<AM_PLAIN, ACT_ELU, 256, 256>(stream, nodeH, 256, semW2, semb2,
                                           nodeA + 128, 384, RN, nullptr, nullptr);
  launch_bn(stream, nodeA + 128, 384, RN, 256, stats, semg, sembt);

  // ---- x_cls = inputs[:,:,:4096] @ clsW + clsb ----
  launch_gemm<AM_PLAIN, ACT_NONE, NODE_T, VISq>(stream, inputs, INW, clsW, clsb,
                                                out_cls, NODE_T, RN, nullptr, nullptr);

  // ---- fusion: relu(nodeA[1280,384] @ fusW + fusb) -> nodeF ----
  launch_gemm<AM_PLAIN, ACT_RELU, 256, 384>(stream, nodeA, 384, fusW, fusb,
                                            nodeF, 256, RN, nullptr, nullptr);

  // ---- m1 MLP on node2edge(nodeF): [101120,512] -> 256 -> 256, BN -> edgeB ----
  launch_gemm<AM_N2E, ACT_ELU, 256, 512>(stream, nullptr, 0, m1W1, m1b1, edgeA,
                                         256, RE, nodeF, nullptr);
  launch_gemm<AM_PLAIN, ACT_ELU, 256, 256>(stream, edgeA, 256, m1W2, m1b2, edgeB,
                                           256, RE, nullptr, nullptr);
  launch_bn(stream, edgeB, 256, RE, 256, stats, m1g, m1bt);   // edgeB = edge_feats

  // ---- edge2node(edgeB) -> nodeG [1280,512] ----
  edge2node_kernel<<<2560, 256, 0, stream>>>(edgeB, nodeG);

  // ---- e2n MLP: [1280,512] -> 256 -> 256, BN -> nodeH2 ----
  launch_gemm<AM_PLAIN, ACT_ELU, 256, 512>(stream, nodeG, 512, e2nW1, e2nb1,
                                           nodeH, 256, RN, nullptr, nullptr);
  launch_gemm<AM_PLAIN, ACT_ELU, 256, 256>(stream, nodeH, 256, e2nW2, e2nb2,
                                           nodeH2, 256, RN, nullptr, nullptr);
  launch_bn(stream, nodeH2, 256, RN, 256, stats, e2ng, e2nbt);

  // ---- m2 MLP: [1280,256] -> 256 -> 256, BN -> nodeF (= h) ----
  launch_gemm<AM_PLAIN, ACT_ELU, 256, 256>(stream, nodeH2, 256, m2W1, m2b1,
                                           nodeH, 256, RN, nullptr, nullptr);
  launch_gemm<AM_PLAIN, ACT_ELU, 256, 256>(stream, nodeH, 256, m2W2, m2b2,
                                           nodeF, 256, RN, nullptr, nullptr);
  launch_bn(stream, nodeF, 256, RN, 256, stats, m2g, m2bt);

  // ---- m3 MLP on [node2edge(h) | edge_feats]: [101120,768] -> 256 -> 256, BN ----
  launch_gemm<AM_M3, ACT_ELU, 256, 768>(stream, nullptr, 0, m3W1, m3b1, edgeA,
                                        256, RE, nodeF, edgeB);
  launch_gemm<AM_PLAIN, ACT_ELU, 256, 256>(stream, edgeA, 256, m3W2, m3b2, edgeB,
                                           256, RE, nullptr, nullptr);
  launch_bn(stream, edgeB, 256, RE, 256, stats, m3g, m3bt);

  // ---- output = edgeB @ relW + relb -> d_out[:,:,:71] ----
  launch_gemm<AM_PLAIN, ACT_NONE, EDGE_T, 256>(stream, edgeB, 256, relW, relb,
                                               out_rel, EDGE_T, RE, nullptr, nullptr);
}